// SparseMeanPool_68247030334366
// MI455X (gfx1250) — compile-verified
//
#include <hip/hip_runtime.h>
#include <stdint.h>

// SparseMeanPool on MI455X (gfx1250):
//   out[b,d,r,c] = (csum[e(r,c)] - csum[s(r,c)]) * scale(r,c)   (0 off-diagonals)
// Store-bandwidth bound: 256 MB out @ 23.3 TB/s ~= 11 us floor.
//  - x rows staged to LDS via CDNA5 async global->LDS DMA (ASYNCcnt).
//  - per-row prefix sums computed as X[32x64] x LowerTriOnes[64x64] on the
//    matrix unit with v_wmma_f32_16x16x4_f32; 8 waves = 8 independent
//    16x16 output blocks (2 row groups x 4 column blocks), no idle waves.
//  - output streamed with coalesced non-temporal b128 stores; (s,e,scale)
//    metadata lives entirely in registers (static function of (r,c)),
//    built once per block and amortized over 32 tiles (512 KB streamed).

#define NTAP 64
#define TILES_PER_WG 32
#define THREADS 256
#define NUM_TILES 16384   // B*D = 32*512

typedef __attribute__((ext_vector_type(2))) float v2f;
typedef __attribute__((ext_vector_type(4))) float v4f;
typedef __attribute__((ext_vector_type(8))) float v8f;

// Static diagonal metadata: replicate _build_maskij([15,8,8], 64) inverted to
// a per-(r,c) lookup. offsets: 1..15 (stride1), 17..31 odd (stride2),
// 35+4m, m=0..7 (stride4). pooled[k] uses s=floor(k*64/L), e=ceil((k+1)*64/L).
__device__ __forceinline__ void diag_entry(int r, int c, int& s, int& e, float& sc) {
  s = 0; e = 0; sc = 0.f;
  const int off = c - r;
  if (off == 0) { s = r; e = r + 1; sc = 1.f; return; }   // main diagonal = x[r]
  int stride;
  if (off >= 1 && off <= 15)                                  stride = 1;
  else if (off >= 17 && off <= 31 && (off & 1))               stride = 2;
  else if (off >= 35 && off <= 63 && (((off - 35) & 3) == 0)) stride = 4;
  else return;
  if (r & (stride - 1)) return;                 // r must be multiple of stride
  const int L = (NTAP - off + stride - 1) / stride;  // len(arange(0, 64-off, stride))
  const int k = r / stride;
  s  = (k * NTAP) / L;
  e  = ((k + 1) * NTAP + L - 1) / L;
  sc = 1.f / (float)(e - s);
}

__global__ __launch_bounds__(THREADS)
void sparse_meanpool_map2d(const float* __restrict__ x, float* __restrict__ out) {
  __shared__ __align__(16) float Xs[TILES_PER_WG][NTAP];       // staged rows (8 KB)
  __shared__ float cs[TILES_PER_WG][NTAP + 1];                 // csum, cs[t][0]=0

  const int tid  = threadIdx.x;
  const int base = blockIdx.x * TILES_PER_WG;

  // ---- per-thread register table for its 16 output positions -------------
  // thread writes float4 chunks at tile-flat offsets 4*(tid + 256*w), w=0..3
  // => r = w*16 + (tid>>4), c = (tid&15)*4 + j   (same for every tile)
  int   S[4][4], E[4][4];
  float SC[4][4];
#pragma unroll
  for (int w = 0; w < 4; ++w)
#pragma unroll
    for (int j = 0; j < 4; ++j) {
      const int r = w * 16 + (tid >> 4);
      const int c = (tid & 15) * 4 + j;
      diag_entry(r, c, S[w][j], E[w][j], SC[w][j]);
    }

  // ---- phase 1: stage 32 rows of x into LDS (async DMA, ASYNCcnt) --------
  // 32 rows x 16 b128-chunks = 512 chunks; 2 per thread.
  {
    const int t = tid >> 4, chunk = tid & 15;
    const float* gp0 = x + (size_t)(base + t) * NTAP + chunk * 4;
    const float* gp1 = gp0 + (size_t)16 * NTAP;          // rows 16..31
#if defined(__gfx1250__)
    const uint32_t l0 = (uint32_t)(uintptr_t)(&Xs[t][chunk * 4]);
    const uint32_t l1 = (uint32_t)(uintptr_t)(&Xs[t + 16][chunk * 4]);
    asm volatile("global_load_async_to_lds_b128 %0, %1, off"
                 :: "v"(l0), "v"((uint64_t)(uintptr_t)gp0) : "memory");
    asm volatile("global_load_async_to_lds_b128 %0, %1, off"
                 :: "v"(l1), "v"((uint64_t)(uintptr_t)gp1) : "memory");
    asm volatile("s_wait_asynccnt 0x0" ::: "memory");
#else
    *(v4f*)&Xs[t][chunk * 4]      = *(const v4f*)gp0;
    *(v4f*)&Xs[t + 16][chunk * 4] = *(const v4f*)gp1;
#endif
  }
  if (tid < TILES_PER_WG) cs[tid][0] = 0.f;
  __syncthreads();

  // ---- phase 2: batched inclusive prefix sums on the matrix unit ---------
  // CS[32 x 64] = Xs[32x64] @ LT[64x64], LT[k][c] = (k <= c).
  // 8 waves = 2 row groups x 4 column blocks; K=64 -> 16 chained 16x16x4 WMMAs.
#if __has_builtin(__builtin_amdgcn_wmma_f32_16x16x4_f32)
  {
    const int wave = tid >> 5;
    const int g16  = (wave >> 2) * 16;     // row group: tiles g16..g16+15
    const int n0   = (wave & 3) * 16;      // column block of this wave
    const int lane = tid & 31;
    const int half = lane >> 4;            // A/B: lanes 16..31 hold K+2,K+3
    const int m    = lane & 15;            // A row (tile idx) / B-D column index
    v8f acc = {0.f, 0.f, 0.f, 0.f, 0.f, 0.f, 0.f, 0.f};
#pragma unroll
    for (int k0 = 0; k0 < NTAP; k0 += 4) {
      const int ka = k0 + (half ? 2 : 0);
      v2f a, b;
      a.x = Xs[g16 + m][ka];
      a.y = Xs[g16 + m][ka + 1];
      const int col = n0 + m;
      b.x = (ka     <= col) ? 1.f : 0.f;   // LT[ka][col]
      b.y = (ka + 1 <= col) ? 1.f : 0.f;   // LT[ka+1][col]
      acc = __builtin_amdgcn_wmma_f32_16x16x4_f32(
          /*neg_a=*/false, a, /*neg_b=*/false, b,
          /*c_mod=*/(short)0, acc, /*reuse_a=*/false, /*reuse_b=*/false);
    }
#pragma unroll
    for (int r = 0; r < 8; ++r)            // D layout: M = r (+8 for hi half)
      cs[g16 + r + (half ? 8 : 0)][1 + n0 + m] = acc[r];
  }
#else
  { // scalar fallback: 8 threads per tile, 8 csum entries each
    const int t = tid >> 3, q = (tid & 7) * 8;
    float acc = 0.f;
    for (int c = 0; c < q + 8; ++c) {
      acc += Xs[t][c];
      if (c >= q) cs[t][c + 1] = acc;
    }
  }
#endif
  __syncthreads();

  // ---- phase 3: stream 32 tiles, coalesced non-temporal b128 stores ------
#pragma unroll 1
  for (int tt = 0; tt < TILES_PER_WG; ++tt) {
    float* op = out + (((size_t)(base + tt)) << 12);   // tile stride 4096
    const float* csr = cs[tt];
#pragma unroll
    for (int w = 0; w < 4; ++w) {
      v4f v;
      v.x = (csr[E[w][0]] - csr[S[w][0]]) * SC[w][0];
      v.y = (csr[E[w][1]] - csr[S[w][1]]) * SC[w][1];
      v.z = (csr[E[w][2]] - csr[S[w][2]]) * SC[w][2];
      v.w = (csr[E[w][3]] - csr[S[w][3]]) * SC[w][3];
      __builtin_nontemporal_store(v, (v4f*)(op + 4 * (tid + THREADS * w)));
    }
  }
}

extern "C" void kernel_launch(void* const* d_in, const int* in_sizes, int n_in,
                              void* d_out, int out_size, void* d_ws, size_t ws_size,
                              hipStream_t stream) {
  (void)in_sizes; (void)n_in; (void)d_ws; (void)ws_size; (void)out_size;
  const float* x = (const float*)d_in[0];
  float* out = (float*)d_out;
  const int blocks = NUM_TILES / TILES_PER_WG;   // 512
  sparse_meanpool_map2d<<<blocks, THREADS, 0, stream>>>(x, out);
}